// TCN_GCN_unit_20710332301733
// MI455X (gfx1250) — compile-verified
//
#include <hip/hip_runtime.h>
#include <math.h>

typedef float v2f __attribute__((ext_vector_type(2)));
typedef float v8f __attribute__((ext_vector_type(8)));

#define N_  256
#define CI  64
#define CO  128
#define T_  64
#define V_  25
#define TQ  32
#define NT  (N_*T_)         // 16384
#define XN  (CI*T_*V_)      // 102400  x n-stride
#define XC  (T_*V_)         // 1600    x c-stride
#define YN  (CO*T_*V_)      // 204800  Y/Z2 n-stride
#define YC  (T_*V_)         // 1600
#define ON  (CO*TQ*V_)      // 102400  out n-stride
#define OC  (TQ*V_)         // 800
#define NBLK 512
#define PA_STRIDE 6912
#define EPSV 1e-5f

// D = A(16x4) * B(4x16) + C, fp32 WMMA
static __device__ __forceinline__ v8f wmma4(float a0, float a1, float b0, float b1, v8f c) {
  v2f a = {a0, a1};
  v2f b = {b0, b1};
  return __builtin_amdgcn_wmma_f32_16x16x4_f32(false, a, false, b, (short)0, c, false, false);
}

// ---------------------------------------------------------------------------
// Pass A: stats for gcn_bn (per (v',d) pre-shift feature), down_bn, res_bn.
// Recomputes gcn/down/res GEMMs with WMMA; nothing stored but partial sums.
// ---------------------------------------------------------------------------
__global__ void __launch_bounds__(256) k_statsA(
    const float* __restrict__ x, const float* __restrict__ Wlin,
    const float* __restrict__ fmask, const float* __restrict__ dw,
    const float* __restrict__ rw, float* __restrict__ pA)
{
  __shared__ float xt[2][64][26];     // 2 (n,t) groups, x[c][v]
  __shared__ float msk[25*64];        // tanh(feat_mask)+1
  __shared__ float gS[3200];          // gcn per-(v',d) sums
  __shared__ float gQ[3200];
  __shared__ float stage[256][4];

  const int tid  = threadIdx.x;
  const int lane = tid & 31, w = tid >> 5;
  const int half = lane >> 4, lr = lane & 15;
  const int dcol = w*16 + lr;

  for (int i = tid; i < 1600; i += 256) msk[i] = tanhf(fmask[i]) + 1.0f;
  for (int i = tid; i < 3200; i += 256) { gS[i] = 0.f; gQ[i] = 0.f; }

  float bg0[16], bg1[16], bd0[16], bd1[16], br0[16], br1[16];
#pragma unroll
  for (int ks = 0; ks < 16; ++ks) {
    int c0 = 4*ks + 2*half, c1 = c0 + 1;
    bg0[ks] = Wlin[c0*128 + dcol]; bg1[ks] = Wlin[c1*128 + dcol];
    bd0[ks] = dw[dcol*64 + c0];    bd1[ks] = dw[dcol*64 + c1];
    br0[ks] = rw[dcol*64 + c0];    br1[ks] = rw[dcol*64 + c1];
  }

  float dS = 0.f, dQ = 0.f, rS = 0.f, rQ = 0.f;
  __syncthreads();

  for (int grp = blockIdx.x; grp < NT/2; grp += gridDim.x) {
    const int nt0 = grp*2;
    for (int i = tid; i < 2*64*25; i += 256) {
      int g = i / 1600, rem = i % 1600;
      int c = rem / 25, v = rem % 25;
      int nt = nt0 + g, n = nt >> 6, t = nt & 63;
      xt[g][c][v] = x[n*XN + c*XC + t*25 + v];
    }
    __syncthreads();

    for (int mt = 0; mt < 4; ++mt) {
      const int rowA = mt*16 + lr;
      const int gA = rowA/25, vA = rowA%25;
      const bool va = rowA < 50;

      // GCN: A[row=(g,v),c] = x[c, (v+c)%25] * mask[v,c]
      v8f Cg = {};
#pragma unroll
      for (int ks = 0; ks < 16; ++ks) {
        int c0 = 4*ks + 2*half, c1 = c0 + 1;
        float a0 = va ? xt[gA][c0][(vA+c0)%25] * msk[vA*64+c0] : 0.f;
        float a1 = va ? xt[gA][c1][(vA+c1)%25] * msk[vA*64+c1] : 0.f;
        Cg = wmma4(a0, a1, bg0[ks], bg1[ks], Cg);
      }
#pragma unroll
      for (int r = 0; r < 8; ++r) {
        int rowC = mt*16 + r + 8*half;
        if (rowC < 50) {                 // addresses race-free by construction
          int vp = rowC % 25;
          float vv = Cg[r];
          gS[vp*128 + dcol] += vv;
          gQ[vp*128 + dcol] += vv*vv;
        }
      }

      // down + res share plain A
      v8f Cd = {}, Cr = {};
#pragma unroll
      for (int ks = 0; ks < 16; ++ks) {
        int c0 = 4*ks + 2*half, c1 = c0 + 1;
        float a0 = va ? xt[gA][c0][vA] : 0.f;
        float a1 = va ? xt[gA][c1][vA] : 0.f;
        Cd = wmma4(a0, a1, bd0[ks], bd1[ks], Cd);
        Cr = wmma4(a0, a1, br0[ks], br1[ks], Cr);
      }
#pragma unroll
      for (int r = 0; r < 8; ++r) {
        int rowC = mt*16 + r + 8*half;
        if (rowC < 50) {
          float vd = Cd[r];
          dS += vd; dQ += vd*vd;
          int nt = nt0 + rowC/25;
          if ((nt & 1) == 0) { float vr = Cr[r]; rS += vr; rQ += vr*vr; }
        }
      }
    }
    __syncthreads();
  }

  for (int i = tid; i < 3200; i += 256) {
    pA[(size_t)blockIdx.x*PA_STRIDE + i]        = gS[i];
    pA[(size_t)blockIdx.x*PA_STRIDE + 3200 + i] = gQ[i];
  }
  stage[tid][0]=dS; stage[tid][1]=dQ; stage[tid][2]=rS; stage[tid][3]=rQ;
  __syncthreads();
  if (tid < 128) {
    int ww = tid >> 4, l = tid & 15;
    int i0 = ww*32 + l, i1 = i0 + 16;
    size_t b = (size_t)blockIdx.x*PA_STRIDE;
    pA[b + 6400 + tid] = stage[i0][0] + stage[i1][0];
    pA[b + 6528 + tid] = stage[i0][1] + stage[i1][1];
    pA[b + 6656 + tid] = stage[i0][2] + stage[i1][2];
    pA[b + 6784 + tid] = stage[i0][3] + stage[i1][3];
  }
}

// ---------------------------------------------------------------------------
__global__ void k_fin1(const float* __restrict__ pA,
                       const float* __restrict__ g1, const float* __restrict__ b1,
                       const float* __restrict__ g2, const float* __restrict__ b2,
                       const float* __restrict__ g5, const float* __restrict__ b5,
                       float* s1, float* o1, float* s2, float* o2,
                       float* s5, float* o5)
{
  int f = blockIdx.x*256 + threadIdx.x;
  if (f >= 3456) return;
  float s = 0.f, q = 0.f;
  if (f < 3200) {
    for (int b = 0; b < NBLK; ++b) {
      s += pA[(size_t)b*PA_STRIDE + f];
      q += pA[(size_t)b*PA_STRIDE + 3200 + f];
    }
    float m = s / 16384.f;
    float var = q / 16384.f - m*m;
    int vp = f / 128, d = f % 128;
    int idx = ((vp + d) % 25)*128 + d;   // post-shift feature index
    float sc = g1[idx] * rsqrtf(var + EPSV);
    s1[idx] = sc; o1[idx] = b1[idx] - m*sc;
  } else if (f < 3328) {
    int d = f - 3200;
    for (int b = 0; b < NBLK; ++b) {
      s += pA[(size_t)b*PA_STRIDE + 6400 + d];
      q += pA[(size_t)b*PA_STRIDE + 6528 + d];
    }
    float m = s / 409600.f; float var = q / 409600.f - m*m;
    float sc = g2[d] * rsqrtf(var + EPSV);
    s2[d] = sc; o2[d] = b2[d] - m*sc;
  } else {
    int d = f - 3328;
    for (int b = 0; b < NBLK; ++b) {
      s += pA[(size_t)b*PA_STRIDE + 6656 + d];
      q += pA[(size_t)b*PA_STRIDE + 6784 + d];
    }
    float m = s / 204800.f; float var = q / 204800.f - m*m;
    float sc = g5[d] * rsqrtf(var + EPSV);
    s5[d] = sc; o5[d] = b5[d] - m*sc;
  }
}

// ---------------------------------------------------------------------------
// Pass B: Y = relu(bn1(out-shifted gcn) + bn(down)), + tcn_bn partial stats.
// ---------------------------------------------------------------------------
__global__ void __launch_bounds__(256) k_passB(
    const float* __restrict__ x, const float* __restrict__ Wlin,
    const float* __restrict__ fmask, const float* __restrict__ dw,
    const float* __restrict__ s1, const float* __restrict__ o1,
    const float* __restrict__ s2, const float* __restrict__ o2,
    float* __restrict__ Y, float* __restrict__ pB)
{
  __shared__ float xt[2][64][26];
  __shared__ float msk[25*64];
  __shared__ float hlds[50*128];      // gcn pre-shift result for this tile
  __shared__ float stage[256][2];

  const int tid  = threadIdx.x;
  const int lane = tid & 31, w = tid >> 5;
  const int half = lane >> 4, lr = lane & 15;
  const int dcol = w*16 + lr;

  for (int i = tid; i < 1600; i += 256) msk[i] = tanhf(fmask[i]) + 1.0f;

  float bg0[16], bg1[16], bd0[16], bd1[16];
#pragma unroll
  for (int ks = 0; ks < 16; ++ks) {
    int c0 = 4*ks + 2*half, c1 = c0 + 1;
    bg0[ks] = Wlin[c0*128 + dcol]; bg1[ks] = Wlin[c1*128 + dcol];
    bd0[ks] = dw[dcol*64 + c0];    bd1[ks] = dw[dcol*64 + c1];
  }
  const float sc2 = s2[dcol], oc2 = o2[dcol];

  float tS = 0.f, tQ = 0.f;
  __syncthreads();

  for (int grp = blockIdx.x; grp < NT/2; grp += gridDim.x) {
    const int nt0 = grp*2;
    for (int i = tid; i < 2*64*25; i += 256) {
      int g = i / 1600, rem = i % 1600;
      int c = rem / 25, v = rem % 25;
      int nt = nt0 + g, n = nt >> 6, t = nt & 63;
      xt[g][c][v] = x[n*XN + c*XC + t*25 + v];
    }
    __syncthreads();

    // phase 1: gcn GEMM -> hlds
    for (int mt = 0; mt < 4; ++mt) {
      const int rowA = mt*16 + lr;
      const int gA = rowA/25, vA = rowA%25;
      const bool va = rowA < 50;
      v8f Cg = {};
#pragma unroll
      for (int ks = 0; ks < 16; ++ks) {
        int c0 = 4*ks + 2*half, c1 = c0 + 1;
        float a0 = va ? xt[gA][c0][(vA+c0)%25] * msk[vA*64+c0] : 0.f;
        float a1 = va ? xt[gA][c1][(vA+c1)%25] * msk[vA*64+c1] : 0.f;
        Cg = wmma4(a0, a1, bg0[ks], bg1[ks], Cg);
      }
#pragma unroll
      for (int r = 0; r < 8; ++r) {
        int rowC = mt*16 + r + 8*half;
        if (rowC < 50) hlds[rowC*128 + dcol] = Cg[r];
      }
    }
    __syncthreads();

    // phase 2: down GEMM + out-shift combine + bn + relu -> Y
    for (int mt = 0; mt < 4; ++mt) {
      const int rowA = mt*16 + lr;
      const int gA = rowA/25, vA = rowA%25;
      const bool va = rowA < 50;
      v8f Cd = {};
#pragma unroll
      for (int ks = 0; ks < 16; ++ks) {
        int c0 = 4*ks + 2*half, c1 = c0 + 1;
        float a0 = va ? xt[gA][c0][vA] : 0.f;
        float a1 = va ? xt[gA][c1][vA] : 0.f;
        Cd = wmma4(a0, a1, bd0[ks], bd1[ks], Cd);
      }
#pragma unroll
      for (int r = 0; r < 8; ++r) {
        int rowC = mt*16 + r + 8*half;
        if (rowC < 50) {
          int g = rowC/25, v = rowC%25;
          int nt = nt0 + g, n = nt >> 6, t = nt & 63;
          int vp = (v - dcol + 150) % 25;       // (v-d) mod 25
          float hv = hlds[(g*25 + vp)*128 + dcol];
          int fidx = v*128 + dcol;
          float yv = s1[fidx]*hv + o1[fidx] + sc2*Cd[r] + oc2;
          yv = fmaxf(yv, 0.f);
          Y[(size_t)n*YN + dcol*YC + t*25 + v] = yv;
          tS += yv; tQ += yv*yv;
        }
      }
    }
    __syncthreads();
  }

  stage[tid][0] = tS; stage[tid][1] = tQ;
  __syncthreads();
  if (tid < 128) {
    int ww = tid >> 4, l = tid & 15;
    int i0 = ww*32 + l, i1 = i0 + 16;
    pB[(size_t)blockIdx.x*256 + tid]       = stage[i0][0] + stage[i1][0];
    pB[(size_t)blockIdx.x*256 + 128 + tid] = stage[i0][1] + stage[i1][1];
  }
}

// ---------------------------------------------------------------------------
__global__ void k_fin2(const float* __restrict__ pB,
                       const float* __restrict__ g3, const float* __restrict__ b3,
                       float* s3, float* o3)
{
  int d = threadIdx.x;
  if (d >= 128) return;
  float s = 0.f, q = 0.f;
  for (int b = 0; b < NBLK; ++b) { s += pB[(size_t)b*256 + d]; q += pB[(size_t)b*256 + 128 + d]; }
  float m = s / 409600.f; float var = q / 409600.f - m*m;
  float sc = g3[d] * rsqrtf(var + EPSV);
  s3[d] = sc; o3[d] = b3[d] - m*sc;
}

// ---------------------------------------------------------------------------
// Pass C: Z2 = relu(tl_w @ shift_in(bn3(Y)) + tl_b). 128x128 WMMA GEMM.
// Block = (n, 4 consecutive t). A tile (100 rows x 128 ch) built in LDS.
// ---------------------------------------------------------------------------
__global__ void __launch_bounds__(256) k_passC(
    const float* __restrict__ Y, const float* __restrict__ tlw,
    const float* __restrict__ tlb, const float* __restrict__ thin,
    const float* __restrict__ s3, const float* __restrict__ o3,
    float* __restrict__ Z2)
{
  __shared__ float At[112*132];
  __shared__ float sc3[128], oc3[128], thf[128];
  __shared__ int   th0[128];

  const int tid  = threadIdx.x;
  const int lane = tid & 31, w = tid >> 5;
  const int half = lane >> 4, lr = lane & 15;
  const int dcol = w*16 + lr;
  const int n  = blockIdx.x >> 4;
  const int t0 = (blockIdx.x & 15) * 4;

  if (tid < 128) {
    float th = thin[tid], fl = floorf(th);
    th0[tid] = (int)fl; thf[tid] = th - fl;
    sc3[tid] = s3[tid]; oc3[tid] = o3[tid];
  }
  __syncthreads();

  // build shifted+bn'd A tile: rows = (tt,v), cols = input channel c
  for (int i = tid; i < 12800; i += 256) {
    int c = i / 100, rowi = i % 100;
    int tt = rowi / 25, v = rowi % 25;
    int ti = t0 + tt + th0[c];
    float f = thf[c];
    const float* yb = Y + (size_t)n*YN + c*YC + v;
    float v0 = (ti   >= 0 && ti   < 64) ? (sc3[c]*yb[ti*25]     + oc3[c]) : 0.f;
    float v1 = (ti+1 >= 0 && ti+1 < 64) ? (sc3[c]*yb[(ti+1)*25] + oc3[c]) : 0.f;
    At[rowi*132 + c] = (1.f - f)*v0 + f*v1;
  }

  float bt0[32], bt1[32];
#pragma unroll
  for (int ks = 0; ks < 32; ++ks) {
    int c0 = 4*ks + 2*half;
    bt0[ks] = tlw[dcol*128 + c0];
    bt1[ks] = tlw[dcol*128 + c0 + 1];
  }
  const float bias = tlb[dcol];
  __syncthreads();

  for (int mt = 0; mt < 7; ++mt) {
    const int rowA = mt*16 + lr;
    const bool va = rowA < 100;
    v8f C = {};
#pragma unroll
    for (int ks = 0; ks < 32; ++ks) {
      int c0 = 4*ks + 2*half;
      float a0 = va ? At[rowA*132 + c0]     : 0.f;
      float a1 = va ? At[rowA*132 + c0 + 1] : 0.f;
      C = wmma4(a0, a1, bt0[ks], bt1[ks], C);
    }
#pragma unroll
    for (int r = 0; r < 8; ++r) {
      int rowC = mt*16 + r + 8*half;
      if (rowC < 100) {
        int tt = rowC/25, v = rowC%25;
        Z2[(size_t)n*YN + dcol*YC + (t0+tt)*25 + v] = fmaxf(C[r] + bias, 0.f);
      }
    }
  }
}

// ---------------------------------------------------------------------------
// Pass D: per-channel stats of shift2(Z2) (deterministic tree reductions).
// ---------------------------------------------------------------------------
__global__ void __launch_bounds__(256) k_statsD(
    const float* __restrict__ Z2, const float* __restrict__ thout,
    float* __restrict__ pD)
{
  __shared__ float red[256][2];
  const int tid = threadIdx.x;
  const int o  = blockIdx.x & 127;
  const int sl = blockIdx.x >> 7;  // 0..15, 16 n's each
  float th = thout[o], fl = floorf(th);
  int d0 = (int)fl; float fr = th - fl;
  float s = 0.f, q = 0.f;
  for (int e = tid; e < 16*32*25; e += 256) {
    int nn = e / 800, rem = e % 800;
    int tq = rem / 25, v = rem % 25;
    int n = sl*16 + nn;
    int ti = 2*tq + d0;
    const float* zb = Z2 + (size_t)n*YN + o*YC + v;
    float v0 = (ti   >= 0 && ti   < 64) ? zb[ti*25]     : 0.f;
    float v1 = (ti+1 >= 0 && ti+1 < 64) ? zb[(ti+1)*25] : 0.f;
    float val = (1.f - fr)*v0 + fr*v1;
    s += val; q += val*val;
  }
  red[tid][0] = s; red[tid][1] = q;
  __syncthreads();
  for (int st = 128; st > 0; st >>= 1) {
    if (tid < st) { red[tid][0] += red[tid+st][0]; red[tid][1] += red[tid+st][1]; }
    __syncthreads();
  }
  if (tid == 0) { pD[(o*16 + sl)*2] = red[0][0]; pD[(o*16 + sl)*2 + 1] = red[0][1]; }
}

__global__ void k_fin4(const float* __restrict__ pD,
                       const float* __restrict__ g4, const float* __restrict__ b4,
                       float* s4, float* o4)
{
  int d = threadIdx.x;
  if (d >= 128) return;
  float s = 0.f, q = 0.f;
  for (int sl = 0; sl < 16; ++sl) { s += pD[(d*16+sl)*2]; q += pD[(d*16+sl)*2 + 1]; }
  float m = s / 204800.f; float var = q / 204800.f - m*m;
  float sc = g4[d] * rsqrtf(var + EPSV);
  s4[d] = sc; o4[d] = b4[d] - m*sc;
}

// ---------------------------------------------------------------------------
// Pass E: out = relu(bn4(shift2(Z2)) + bn5(res GEMM)).
// ---------------------------------------------------------------------------
__global__ void __launch_bounds__(256) k_passE(
    const float* __restrict__ x, const float* __restrict__ rw,
    const float* __restrict__ thout,
    const float* __restrict__ s4, const float* __restrict__ o4,
    const float* __restrict__ s5, const float* __restrict__ o5,
    const float* __restrict__ Z2, float* __restrict__ out)
{
  __shared__ float xt[2][64][26];

  const int tid  = threadIdx.x;
  const int lane = tid & 31, w = tid >> 5;
  const int half = lane >> 4, lr = lane & 15;
  const int dcol = w*16 + lr;

  float th = thout[dcol], fl = floorf(th);
  const int d0 = (int)fl; const float fr = th - fl;
  const float sc4 = s4[dcol], os4 = o4[dcol];
  const float sc5 = s5[dcol], os5 = o5[dcol];

  float br0[16], br1[16];
#pragma unroll
  for (int ks = 0; ks < 16; ++ks) {
    int c0 = 4*ks + 2*half;
    br0[ks] = rw[dcol*64 + c0];
    br1[ks] = rw[dcol*64 + c0 + 1];
  }

  for (int grp = blockIdx.x; grp < (N_*TQ)/2; grp += gridDim.x) {
    const int ntq0 = grp*2;
    for (int i = tid; i < 2*64*25; i += 256) {
      int g = i / 1600, rem = i % 1600;
      int c = rem / 25, v = rem % 25;
      int ntq = ntq0 + g, n = ntq >> 5, tq = ntq & 31;
      xt[g][c][v] = x[n*XN + c*XC + (2*tq)*25 + v];
    }
    __syncthreads();

    for (int mt = 0; mt < 4; ++mt) {
      const int rowA = mt*16 + lr;
      const int gA = rowA/25, vA = rowA%25;
      const bool va = rowA < 50;
      v8f Cr = {};
#pragma unroll
      for (int ks = 0; ks < 16; ++ks) {
        int c0 = 4*ks + 2*half, c1 = c0 + 1;
        float a0 = va ? xt[gA][c0][vA] : 0.f;
        float a1 = va ? xt[gA][c1][vA] : 0.f;
        Cr = wmma4(a0, a1, br0[ks], br1[ks], Cr);
      }
#pragma unroll
      for (int r = 0; r < 8; ++r) {
        int rowC = mt*16 + r + 8*half;
        if (rowC < 50) {
          int g = rowC/25, v = rowC%25;
          int ntq = ntq0 + g, n = ntq >> 5, tq = ntq & 31;
          int ti = 2*tq + d0;
          const float* zb = Z2 + (size_t)n*YN + dcol*YC + v;
          float v0 = (ti   >= 0 && ti   < 64) ? zb[ti*25]     : 0.f;
          float v1 = (ti+1 >= 0 && ti+1 < 64) ? zb[(ti+1)*25] : 0.f;
          float zs = (1.f - fr)*v0 + fr*v1;
          float val = fmaxf(sc4*zs + os4 + sc5*Cr[r] + os5, 0.f);
          out[(size_t)n*ON + dcol*OC + tq*25 + v] = val;
        }
      }
    }
    __syncthreads();
  }
}

// ---------------------------------------------------------------------------
extern "C" void kernel_launch(void* const* d_in, const int* in_sizes, int n_in,
                              void* d_out, int out_size, void* d_ws, size_t ws_size,
                              hipStream_t stream)
{
  const float* x     = (const float*)d_in[0];
  const float* Wlin  = (const float*)d_in[1];
  // d_in[2] b_lin: exact no-op through bn1d
  const float* fmask = (const float*)d_in[3];
  const float* g1 = (const float*)d_in[4];
  const float* b1 = (const float*)d_in[5];
  const float* dw = (const float*)d_in[6];
  // d_in[7] down_b: no-op through down_bn
  const float* g2 = (const float*)d_in[8];
  const float* b2 = (const float*)d_in[9];
  const float* g3 = (const float*)d_in[10];
  const float* b3 = (const float*)d_in[11];
  const float* thin  = (const float*)d_in[12];
  const float* tlw   = (const float*)d_in[13];
  const float* tlb   = (const float*)d_in[14];
  const float* thout = (const float*)d_in[15];
  const float* g4 = (const float*)d_in[16];
  const float* b4 = (const float*)d_in[17];
  const float* rw = (const float*)d_in[18];
  // d_in[19] res_b: no-op through res_bn
  const float* g5 = (const float*)d_in[20];
  const float* b5 = (const float*)d_in[21];

  float* ws = (float*)d_ws;
  float* Y  = ws;
  float* Z2 = Y  + (size_t)N_*CO*T_*V_;
  float* pA = Z2 + (size_t)N_*CO*T_*V_;
  float* pB = pA + (size_t)NBLK*PA_STRIDE;
  float* pD = pB + (size_t)NBLK*256;
  float* s1 = pD + 4096;
  float* o1 = s1 + 3200;
  float* s2 = o1 + 3200;
  float* o2 = s2 + 128;
  float* s3 = o2 + 128;
  float* o3 = s3 + 128;
  float* s4 = o3 + 128;
  float* o4 = s4 + 128;
  float* s5 = o4 + 128;
  float* o5 = s5 + 128;

  k_statsA<<<NBLK, 256, 0, stream>>>(x, Wlin, fmask, dw, rw, pA);
  k_fin1  <<<14, 256, 0, stream>>>(pA, g1, b1, g2, b2, g5, b5, s1, o1, s2, o2, s5, o5);
  k_passB <<<NBLK, 256, 0, stream>>>(x, Wlin, fmask, dw, s1, o1, s2, o2, Y, pB);
  k_fin2  <<<1, 128, 0, stream>>>(pB, g3, b3, s3, o3);
  k_passC <<<N_*16, 256, 0, stream>>>(Y, tlw, tlb, thin, s3, o3, Z2);
  k_statsD<<<128*16, 256, 0, stream>>>(Z2, thout, pD);
  k_fin4  <<<1, 128, 0, stream>>>(pD, g4, b4, s4, o4);
  k_passE <<<NBLK, 256, 0, stream>>>(x, rw, thout, s4, o4, s5, o5, Z2, (float*)d_out);
}